// NodeBlock_2929167696135
// MI455X (gfx1250) — compile-verified
//
#include <hip/hip_runtime.h>

typedef float v2f __attribute__((ext_vector_type(2)));
typedef float v4f __attribute__((ext_vector_type(4)));
typedef float v8f __attribute__((ext_vector_type(8)));

#define N_NODES   50000
#define N_EDGES   800000
#define H         128
#define DIN       256
#define DOUT      256
#define TILE_M    128
#define N_TILES   (N_EDGES / TILE_M)   // 6250 exact
#define KC        32
#define GEMM_BLOCKS 1250               // 5 tiles per block
#define LPAD      36                   // stride%64=36 (gcd 4 -> 16 distinct banks), 16B-aligned rows

// Use CDNA5 async global->LDS staging (ASYNCcnt path). Flip to 0 if assembler rejects.
#define USE_ASYNC_STAGE 1

// d_ws float-offset layout
#define WS_SCALE1 0                    // 256
#define WS_SHIFT1 256                  // 256
#define WS_SUM2   512                  // 128
#define WS_SSQ2   640                  // 128
#define WS_SCALE2 768                  // 128
#define WS_SHIFT2 896                  // 128
#define WS_PART1  1024                 // GEMM_BLOCKS * 512 partial col stats

__device__ __forceinline__ void cp16(float* dst, const float* src) {
#if USE_ASYNC_STAGE
  // LDS byte offset = low 32 bits of the flat shared pointer (ISA 10.2: LDS aperture
  // keeps the wave-relative offset in addr[31:0]); hardware adds LDS_BASE.
  asm volatile("global_load_async_to_lds_b128 %0, %1, off"
               :
               : "v"((unsigned int)(unsigned long long)(uintptr_t)dst),
                 "v"((unsigned long long)(uintptr_t)src)
               : "memory");
#else
  *(v4f*)dst = *(const v4f*)src;
#endif
}

__device__ __forceinline__ void staging_wait() {
#if USE_ASYNC_STAGE
  asm volatile("s_wait_asynccnt 0x0" ::: "memory");
#endif
}

// ---------------- zero init: node accumulator (d_out) + node-stat slots ----------------
__global__ void zero_kernel(float* __restrict__ out, float* __restrict__ ws) {
  size_t i = (size_t)blockIdx.x * blockDim.x + threadIdx.x;
  if (i < (size_t)N_NODES * H) {
    out[i] = 0.0f;
  } else if (i < (size_t)N_NODES * H + 256) {
    ws[WS_SUM2 + (i - (size_t)N_NODES * H)] = 0.0f;
  }
}

// ---------------- edge GEMM: y = concat(node[i],edge) @ W1^T (+b1) ----------------
// Wave w = g*2+(w&1) layout: mg (w&1) owns 64 rows, g (w>>1) owns paired
// filter cols [g*32,g*32+32) AND core cols +128 -> gate needs no LDS exchange.
template <bool STATS>
__global__ __launch_bounds__(256) void edge_gemm_kernel(
    const float* __restrict__ node_emb, const float* __restrict__ edge_emb,
    const int* __restrict__ eidx, const float* __restrict__ w1,
    const float* __restrict__ b1, float* __restrict__ ws,
    float* __restrict__ nacc) {
  __shared__ float xs[TILE_M * LPAD];   // 18432 B : x k-chunk
  __shared__ float wl[DOUT * LPAD];     // 36864 B : w1 k-chunk
  __shared__ int   si[TILE_M];          // gathered node ids
  __shared__ float redS[256];
  __shared__ float redQ[256];

  const int tid  = threadIdx.x;
  const int wave = tid >> 5;
  const int lane = tid & 31;
  const int l16  = lane & 15;
  const int lhi  = lane >> 4;
  const int mg   = (wave & 1) * 64;  // 2 m-groups x 64 rows
  const int g    = wave >> 1;        // 4 col-groups x (32 filter + 32 core)

  // per-lane columns: i = j*2+p  (j: 16-col subtile, p: 0=filter,1=core)
  int cols[4];
#pragma unroll
  for (int j = 0; j < 2; ++j) {
    cols[j * 2 + 0] = g * 32 + j * 16 + l16;
    cols[j * 2 + 1] = g * 32 + j * 16 + l16 + 128;
  }

  float biasv[4], scv[4], shv[4];
#pragma unroll
  for (int i = 0; i < 4; ++i) biasv[i] = b1[cols[i]];
  if (!STATS) {
#pragma unroll
    for (int i = 0; i < 4; ++i) {
      scv[i] = ws[WS_SCALE1 + cols[i]];
      shv[i] = ws[WS_SHIFT1 + cols[i]] + biasv[i] * scv[i];  // fold bias
    }
  }

  float sS[4] = {0.f, 0.f, 0.f, 0.f};
  float sQ[4] = {0.f, 0.f, 0.f, 0.f};

  const int xrow  = tid >> 1;        // staging: 128 rows, 2 threads/row
  const int xcol0 = (tid & 1) * 16;  // 16 floats each

  for (int t = blockIdx.x; t < N_TILES; t += gridDim.x) {
    const int e0 = t * TILE_M;
    if (tid < TILE_M) si[tid] = eidx[e0 + tid];

    v8f acc[16];
#pragma unroll
    for (int i = 0; i < 16; ++i) acc[i] = (v8f)0.0f;

    for (int kc = 0; kc < DIN; kc += KC) {
      __syncthreads();  // si visible (1st iter) / LDS safe to overwrite
      {   // stage x chunk: rows 0..127, cols kc..kc+31
        const float* src = (kc < H)
            ? node_emb + (size_t)si[xrow] * H + kc + xcol0
            : edge_emb + (size_t)(e0 + xrow) * H + (kc - H) + xcol0;
        float* dst = &xs[xrow * LPAD + xcol0];
#pragma unroll
        for (int q = 0; q < 4; ++q) cp16(dst + q * 4, src + q * 4);
      }
      {   // stage w1 chunk: row n=tid, cols kc..kc+31
        const float* wsrc = w1 + (size_t)tid * DIN + kc;
        float* wdst = &wl[tid * LPAD];
#pragma unroll
        for (int q = 0; q < 8; ++q) cp16(wdst + q * 4, wsrc + q * 4);
      }
      staging_wait();
      __syncthreads();
#pragma unroll
      for (int k0 = 0; k0 < KC; k0 += 4) {
        const int kk = k0 + 2 * lhi;  // K pair per ISA f32 A/B layout
        v2f a[4], bb[4];
#pragma unroll
        for (int ms = 0; ms < 4; ++ms)
          a[ms] = *(const v2f*)&xs[(mg + ms * 16 + l16) * LPAD + kk];
#pragma unroll
        for (int i = 0; i < 4; ++i)
          bb[i] = *(const v2f*)&wl[cols[i] * LPAD + kk];
#pragma unroll
        for (int ms = 0; ms < 4; ++ms)
#pragma unroll
          for (int i = 0; i < 4; ++i)
            acc[ms * 4 + i] = __builtin_amdgcn_wmma_f32_16x16x4_f32(
                false, a[ms], false, bb[i], (short)0, acc[ms * 4 + i], false, false);
      }
    }

    if (STATS) {
#pragma unroll
      for (int i = 0; i < 4; ++i) {
        float ss = 0.f, qq = 0.f;
#pragma unroll
        for (int ms = 0; ms < 4; ++ms)
#pragma unroll
          for (int v = 0; v < 8; ++v) {
            const float y = acc[ms * 4 + i][v] + biasv[i];
            ss += y;
            qq += y * y;
          }
        sS[i] += ss;
        sQ[i] += qq;
      }
    } else {
      // gate + scatter straight from registers (filter/core share lane+slot)
#pragma unroll
      for (int ms = 0; ms < 4; ++ms) {
#pragma unroll
        for (int v = 0; v < 8; ++v) {
          const int m = mg + ms * 16 + v + 8 * lhi;  // D layout: vgpr v -> M=v / v+8
          float* ob = nacc + (size_t)si[m] * H + g * 32 + l16;
#pragma unroll
          for (int j = 0; j < 2; ++j) {
            const float yf = acc[ms * 4 + j * 2 + 0][v] * scv[j * 2 + 0] + shv[j * 2 + 0];
            const float yc = acc[ms * 4 + j * 2 + 1][v] * scv[j * 2 + 1] + shv[j * 2 + 1];
            const float msg = (1.0f / (1.0f + __expf(-yf))) * tanhf(yc);
            atomicAdd(ob + j * 16, msg);
          }
        }
      }
    }
    __syncthreads();  // si/LDS reused next tile
  }

  if (STATS) {
    redS[tid] = 0.f;
    redQ[tid] = 0.f;
    __syncthreads();
#pragma unroll
    for (int i = 0; i < 4; ++i) {
      atomicAdd(&redS[cols[i]], sS[i]);
      atomicAdd(&redQ[cols[i]], sQ[i]);
    }
    __syncthreads();
    float* part = ws + WS_PART1 + (size_t)blockIdx.x * 512;
    part[tid]       = redS[tid];
    part[256 + tid] = redQ[tid];
  }
}

// ---------------- BN1 params ----------------
__global__ void bn1_params_kernel(const float* __restrict__ g1,
                                  const float* __restrict__ be1,
                                  float* __restrict__ ws) {
  const int j = threadIdx.x;  // 256 threads
  float s = 0.f, q = 0.f;
  for (int b = 0; b < GEMM_BLOCKS; ++b) {
    const float* part = ws + WS_PART1 + (size_t)b * 512;
    s += part[j];
    q += part[256 + j];
  }
  const float inv = 1.0f / (float)N_EDGES;
  const float mean = s * inv;
  const float var  = q * inv - mean * mean;
  const float scale = g1[j] * rsqrtf(var + 1e-5f);
  ws[WS_SCALE1 + j] = scale;
  ws[WS_SHIFT1 + j] = be1[j] - mean * scale;
}

// ---------------- node column stats over accumulator [50000][128] ----------------
__global__ void node_stats_kernel(const float* __restrict__ nacc,
                                  float* __restrict__ ws) {
  const int c   = threadIdx.x & 127;
  const int sub = threadIdx.x >> 7;
  const int r0  = blockIdx.x * 200;  // 250 blocks * 200 rows
  float s = 0.f, q = 0.f;
  for (int r = r0 + sub; r < r0 + 200; r += 2) {
    const float v = nacc[(size_t)r * H + c];
    s += v;
    q += v * v;
  }
  atomicAdd(&ws[WS_SUM2 + c], s);
  atomicAdd(&ws[WS_SSQ2 + c], q);
}

__global__ void bn2_params_kernel(const float* __restrict__ g2,
                                  const float* __restrict__ be2,
                                  float* __restrict__ ws) {
  const int c = threadIdx.x;  // 128 threads
  const float inv = 1.0f / (float)N_NODES;
  const float mean = ws[WS_SUM2 + c] * inv;
  const float var  = ws[WS_SSQ2 + c] * inv - mean * mean;
  const float scale = g2[c] * rsqrtf(var + 1e-5f);
  ws[WS_SCALE2 + c] = scale;
  ws[WS_SHIFT2 + c] = be2[c] - mean * scale;
}

// ---------------- out = tanh(node_emb + bn2(acc)) ----------------
__global__ void final_kernel(const float* __restrict__ node_emb,
                             float* __restrict__ out,
                             const float* __restrict__ ws) {
  const size_t i4 = (size_t)blockIdx.x * blockDim.x + threadIdx.x;
  if (i4 >= (size_t)N_NODES * H / 4) return;
  const int c0 = (int)((i4 * 4) & (H - 1));
  v4f ne = ((const v4f*)node_emb)[i4];
  v4f ac = ((v4f*)out)[i4];
  v4f scvv = *(const v4f*)&ws[WS_SCALE2 + c0];
  v4f shvv = *(const v4f*)&ws[WS_SHIFT2 + c0];
  v4f r;
#pragma unroll
  for (int q = 0; q < 4; ++q) r[q] = tanhf(ne[q] + ac[q] * scvv[q] + shvv[q]);
  ((v4f*)out)[i4] = r;
}

extern "C" void kernel_launch(void* const* d_in, const int* in_sizes, int n_in,
                              void* d_out, int out_size, void* d_ws, size_t ws_size,
                              hipStream_t stream) {
  const float* node_emb = (const float*)d_in[0];
  const float* edge_emb = (const float*)d_in[1];
  const int*   eidx     = (const int*)d_in[2];
  const float* w1       = (const float*)d_in[3];
  const float* b1       = (const float*)d_in[4];
  const float* g1       = (const float*)d_in[5];
  const float* be1      = (const float*)d_in[6];
  const float* g2       = (const float*)d_in[7];
  const float* be2      = (const float*)d_in[8];
  float* out = (float*)d_out;
  float* ws  = (float*)d_ws;
  (void)in_sizes; (void)n_in; (void)out_size; (void)ws_size;

  const size_t nz = (size_t)N_NODES * H + 256;
  zero_kernel<<<(unsigned)((nz + 255) / 256), 256, 0, stream>>>(out, ws);

  edge_gemm_kernel<true><<<GEMM_BLOCKS, 256, 0, stream>>>(
      node_emb, edge_emb, eidx, w1, b1, ws, out);
  bn1_params_kernel<<<1, 256, 0, stream>>>(g1, be1, ws);
  edge_gemm_kernel<false><<<GEMM_BLOCKS, 256, 0, stream>>>(
      node_emb, edge_emb, eidx, w1, b1, ws, out);

  node_stats_kernel<<<250, 256, 0, stream>>>(out, ws);
  bn2_params_kernel<<<1, 128, 0, stream>>>(g2, be2, ws);

  final_kernel<<<(unsigned)((N_NODES * H / 4 + 255) / 256), 256, 0, stream>>>(
      node_emb, out, ws);
}